// RAFTStereoFusion_25975962206437
// MI455X (gfx1250) — compile-verified
//
#include <hip/hip_runtime.h>
#include <hip/hip_bf16.h>
#include <stdint.h>

// Problem geometry (fixed by setup_inputs)
#define N_   8
#define C_   256
#define H_   192
#define W_   320
#define HW_  (H_ * W_)
#define FH_  768
#define FW_  1280
#define FHW_ (FH_ * FW_)
#define MCH_ 144        // 9 * FACTOR * FACTOR
#define EPS_ 1e-6f

typedef __attribute__((ext_vector_type(4))) unsigned int v4u;
typedef __attribute__((ext_vector_type(8))) int          v8i;
typedef __attribute__((ext_vector_type(4))) int          v4i;

__device__ __forceinline__ int imin_(int a, int b) { return a < b ? a : b; }
__device__ __forceinline__ int imax_(int a, int b) { return a > b ? a : b; }

// value of F.interpolate(img[Hin,Win] -> [oh,ow], bilinear, align_corners=False) at (y,x)
__device__ __forceinline__ float resize_at(const float* __restrict__ img,
                                           int Hin, int Win, int oh, int ow,
                                           int y, int x) {
  float ys = ((float)y + 0.5f) * ((float)Hin / (float)oh) - 0.5f;
  float xs = ((float)x + 0.5f) * ((float)Win / (float)ow) - 0.5f;
  float y0f = floorf(ys), x0f = floorf(xs);
  float wy = ys - y0f, wx = xs - x0f;
  int y0 = imin_(imax_((int)y0f,     0), Hin - 1);
  int y1 = imin_(imax_((int)y0f + 1, 0), Hin - 1);
  int x0 = imin_(imax_((int)x0f,     0), Win - 1);
  int x1 = imin_(imax_((int)x0f + 1, 0), Win - 1);
  float t = img[(size_t)y0 * Win + x0] * (1.f - wx) + img[(size_t)y0 * Win + x1] * wx;
  float b = img[(size_t)y1 * Win + x0] * (1.f - wx) + img[(size_t)y1 * Win + x1] * wx;
  return t * (1.f - wy) + b * wy;
}

// ---------------------------------------------------------------------------
// Kernel 1: warped-feature fusion.  1 thread = 1 (n,h,w) pixel, loops channels.
// Per-pixel params computed once, amortized over 256 coalesced channel rows.
// ---------------------------------------------------------------------------
__global__ __launch_bounds__(256) void fuse_kernel(
    const float* __restrict__ fmap1, const float* __restrict__ fmap1n,
    const float* __restrict__ flowL, const float* __restrict__ mask1,
    const float* __restrict__ mask1n, float* __restrict__ out) {
  int p = blockIdx.x * 256 + threadIdx.x;
  if (p >= N_ * HW_) return;
  int n  = p / HW_;
  int hw = p - n * HW_;
  int h  = hw / W_;
  int w  = hw - h * W_;

  // resize flow (FHxFW -> HxW) at (h,w), then component scaling
  const float* flx = flowL + (size_t)(n * 2) * FHW_;
  const float* fly = flx + FHW_;
  float rfx = resize_at(flx, FH_, FW_, H_, W_, h, w) * ((float)W_ / (float)FW_);
  float rfy = resize_at(fly, FH_, FW_, H_, W_, h, w) * ((float)H_ / (float)FH_);

  // grid build + grid_sample pixel coords (align_corners=False)
  float gxn = 2.f * ((float)w + rfx) / (float)(W_ - 1) - 1.f;
  float gyn = 2.f * ((float)h + rfy) / (float)(H_ - 1) - 1.f;
  float px = ((gxn + 1.f) * (float)W_ - 1.f) * 0.5f;
  float py = ((gyn + 1.f) * (float)H_ - 1.f) * 0.5f;
  float fx0 = floorf(px), fy0 = floorf(py);
  int ix0 = (int)fx0, iy0 = (int)fy0;
  float ax = px - fx0, ay = py - fy0;

  float wt[4] = { (1.f - ax) * (1.f - ay), ax * (1.f - ay),
                  (1.f - ax) * ay,         ax * ay };
  int ty[4] = { iy0, iy0, iy0 + 1, iy0 + 1 };
  int tx[4] = { ix0, ix0 + 1, ix0, ix0 + 1 };

  // warped resized mask1_next (zeros padding) + tap weights for feature gather
  const float* m1n = mask1n + (size_t)n * FHW_;
  float rm1n = 0.f;
  int   idx[4];
  float bw[4];
#pragma unroll
  for (int k = 0; k < 4; ++k) {
    bool v = (ty[k] >= 0) && (ty[k] < H_) && (tx[k] >= 0) && (tx[k] < W_);
    int yc = imin_(imax_(ty[k], 0), H_ - 1);
    int xc = imin_(imax_(tx[k], 0), W_ - 1);
    idx[k] = yc * W_ + xc;
    float mv = v ? resize_at(m1n, FH_, FW_, H_, W_, yc, xc) : 0.f;
    rm1n += wt[k] * mv;
    bw[k] = v ? wt[k] : 0.f;
  }

  float rm1 = resize_at(mask1 + (size_t)n * FHW_, FH_, FW_, H_, W_, h, w);
  float s = rm1 + rm1n;
  if (s == 0.f) s = EPS_;
  float inv = 1.f / s;
  float a  = rm1 * inv;
  float cb = rm1n * inv;
  float b0 = cb * bw[0], b1 = cb * bw[1], b2 = cb * bw[2], b3 = cb * bw[3];

  const float* f1 = fmap1  + (size_t)n * C_ * HW_ + hw;
  const float* f2 = fmap1n + (size_t)n * C_ * HW_;
  float*       fo = out    + (size_t)n * C_ * HW_ + hw;

#pragma unroll 4
  for (int c = 0; c < C_; ++c) {
    size_t o = (size_t)c * HW_;
    float v1 = __builtin_nontemporal_load(f1 + o);           // read-once stream
    float g = b0 * f2[o + idx[0]] + b1 * f2[o + idx[1]]
            + b2 * f2[o + idx[2]] + b3 * f2[o + idx[3]];     // gather, keep cached
    __builtin_nontemporal_store(a * v1 + g, fo + o);         // write-once stream
  }
}

// ---------------------------------------------------------------------------
// Kernel 2: convex-combination flow upsampling (keep x/disparity channel).
// Block = 32x8 coarse pixels.  Coarse-flow tile (10x34 incl. halo) is staged
// into LDS via the Tensor Data Mover; TDM OOB-zero fill reproduces the
// reference's zero padding on the right/bottom; software zero for -1 halo.
// ---------------------------------------------------------------------------
__global__ __launch_bounds__(256) void upsample_kernel(
    const float* __restrict__ cflow, const float* __restrict__ upmask,
    float* __restrict__ oflow) {
  __shared__ float lflow[10 * 34];

  int n = blockIdx.z, by = blockIdx.y, bx = blockIdx.x;
  int gh0 = by * 8, gw0 = bx * 32;
  int sy = imax_(gh0 - 1, 0);     // tile start (clamped for top/left halo)
  int sx = imax_(gw0 - 1, 0);

  // defensive zero of the staging tile (covers -1 halo slots / OOB semantics)
  for (int i = threadIdx.x; i < 10 * 34; i += 256) lflow[i] = 0.f;
  __syncthreads();

  if (threadIdx.x == 0) {
    // Tensor DMA descriptor (D#): 2D tile of flow channel 0, data_size = 4B.
    unsigned long long ga = (unsigned long long)(uintptr_t)(
        cflow + ((size_t)(n * 2) * H_ + sy) * W_ + sx);
    unsigned ldsoff = (unsigned)(uintptr_t)(void*)lflow;  // low 32b = LDS offset
    unsigned td0 = (unsigned)(W_ - sx);  // cols beyond -> zero-fill (right pad)
    unsigned td1 = (unsigned)(H_ - sy);  // rows beyond -> zero-fill (bottom pad)

    v4u g0;
    g0[0] = 1u;                                   // count=1, user descriptor
    g0[1] = ldsoff;                               // lds_addr (bytes)
    g0[2] = (unsigned)(ga & 0xFFFFFFFFull);       // global_addr[31:0]
    g0[3] = (unsigned)((ga >> 32) & 0x01FFFFFFull) | (2u << 30);  // [56:32] | type=2

    v8i g1;
    g1[0] = (int)(2u << 16);                                        // data_size=4B
    g1[1] = (int)((td0 & 0xFFFFu) << 16);                           // tensor_dim0 lo
    g1[2] = (int)(((td0 >> 16) & 0xFFFFu) | ((td1 & 0xFFFFu) << 16));
    g1[3] = (int)(((td1 >> 16) & 0xFFFFu) | (34u << 16));           // tile_dim0=34
    g1[4] = (int)10u;                                               // tile_dim1=10
    g1[5] = (int)W_;                                                // dim0 stride
    g1[6] = 0;
    g1[7] = 0;

    v4i g2 = {0, 0, 0, 0};
    v4i g3 = {0, 0, 0, 0};
    v8i g4 = {0, 0, 0, 0, 0, 0, 0, 0};   // extra descriptor words (6-arg toolchain)
    __builtin_amdgcn_tensor_load_to_lds(g0, g1, g2, g3, g4, 0);
  }
  __builtin_amdgcn_s_wait_tensorcnt(0);
  __syncthreads();

  int tw = threadIdx.x & 31;       // lane = w  -> 128B-contiguous mask reads
  int th = threadIdx.x >> 5;
  int h = gh0 + th, w = gw0 + tw;

  // 3x3 zero-padded neighborhood of FACTOR*flow, row-major (unfold order)
  float nb[9];
#pragma unroll
  for (int dy = 0; dy < 3; ++dy)
#pragma unroll
    for (int dx = 0; dx < 3; ++dx) {
      int ry = (h - 1 + dy) - sy;
      int rx = (w - 1 + dx) - sx;
      float v = (ry >= 0 && rx >= 0) ? lflow[ry * 34 + rx] : 0.f;
      nb[dy * 3 + dx] = 4.0f * v;
    }

  const float* mbase = upmask + (size_t)n * MCH_ * HW_ + (size_t)h * W_ + w;

#pragma unroll 1
  for (int f = 0; f < 16; ++f) {      // f = fy*4 + fx
    if (f < 15) __builtin_prefetch(mbase + (size_t)(f + 1) * HW_, 0, 1);
    float m[9], mx = -3.4e38f;
#pragma unroll
    for (int k = 0; k < 9; ++k) {     // mask channel = k*16 + f
      m[k] = mbase[(size_t)(k * 16 + f) * HW_];
      mx = fmaxf(mx, m[k]);
    }
    float sum = 0.f, acc = 0.f;
#pragma unroll
    for (int k = 0; k < 9; ++k) {
      float e = __expf(m[k] - mx);    // softmax over the 9 neighbors
      sum += e;
      acc += e * nb[k];
    }
    int Y = 4 * h + (f >> 2);
    int X = 4 * w + (f & 3);
    oflow[((size_t)n * FH_ + Y) * FW_ + X] = acc / sum;
  }
}

// ---------------------------------------------------------------------------
extern "C" void kernel_launch(void* const* d_in, const int* in_sizes, int n_in,
                              void* d_out, int out_size, void* d_ws, size_t ws_size,
                              hipStream_t stream) {
  (void)in_sizes; (void)n_in; (void)d_ws; (void)ws_size; (void)out_size;
  const float* fmap1  = (const float*)d_in[0];
  const float* fmap1n = (const float*)d_in[1];
  const float* flowL  = (const float*)d_in[2];
  const float* mask1  = (const float*)d_in[3];
  const float* mask1n = (const float*)d_in[4];
  const float* cflow  = (const float*)d_in[5];
  const float* upmask = (const float*)d_in[6];

  float* out_fused = (float*)d_out;                               // [N,C,H,W]
  float* out_flow  = out_fused + (size_t)N_ * C_ * HW_;           // [N,1,FH,FW]

  fuse_kernel<<<dim3((N_ * HW_) / 256), dim3(256), 0, stream>>>(
      fmap1, fmap1n, flowL, mask1, mask1n, out_fused);

  upsample_kernel<<<dim3(W_ / 32, H_ / 8, N_), dim3(256), 0, stream>>>(
      cflow, upmask, out_flow);
}